// Kraskov_critic_3736621548241
// MI455X (gfx1250) — compile-verified
//
#include <hip/hip_runtime.h>
#include <math.h>

#define NPTS 8192
#define DXD 16
#define DYD 16
#define TILE_J 64                    // j-rows per tile (per buffer)
#define NBUF 2                       // double buffering
#define WAVES 8
#define ROWS 32                      // rows per block (== lanes per wave)
#define THREADS 256
#define JCHUNK (NPTS / WAVES)        // 1024 j's per wave
#define TILES_PER_CHUNK (JCHUNK / TILE_J)   // 16 tiles
#define ASY_PER_TILE 16              // 2 arrays x 4KB / (32 lanes x 16B) = 8+8 issues

// ---------- CDNA5 async-tensor path: global -> LDS without VGPR round trip ----
__device__ __forceinline__ void async_ld_b128(uint32_t lds_off, const void* g) {
  // GV mode: vdst = LDS byte address, vaddr = 64-bit global address
  asm volatile("global_load_async_to_lds_b128 %0, %1, off"
               :: "v"(lds_off), "v"(g) : "memory");
}
__device__ __forceinline__ void wait_async0() {   // all copies landed
  asm volatile("s_wait_asynccnt 0" ::: "memory");
}
__device__ __forceinline__ void wait_async_prev() {
  // <= ASY_PER_TILE outstanding: previous tile complete (async loads retire in order),
  // current tile's copies still in flight behind the compute.
  asm volatile("s_wait_asynccnt 16" ::: "memory");
}
__device__ __forceinline__ void wait_ds0() {      // drain LDS reads before buffer reuse (WAR)
  asm volatile("s_wait_dscnt 0" ::: "memory");
}

// copy 4KB (64 rows x 16 floats, contiguous in global) into wave-private LDS
__device__ __forceinline__ void wave_async_copy_4k(float* ldsdst, const float* gsrc, int lane) {
  uint32_t base = (uint32_t)(uintptr_t)ldsdst;    // low 32 bits of generic ptr = LDS byte offset
  const char* s = (const char*)gsrc;
#pragma unroll
  for (int rnd = 0; rnd < 8; ++rnd) {             // 8 rounds x 32 lanes x 16B = 4KB
    uint32_t off = (uint32_t)((rnd * 32 + lane) * 16);
    async_ld_b128(base + off, s + off);
  }
}

// ---------- math helpers ------------------------------------------------------
__device__ __forceinline__ float linf4(float4 a, float4 b) {
  float m0 = fmaxf(fabsf(a.x - b.x), fabsf(a.y - b.y));
  float m1 = fmaxf(fabsf(a.z - b.z), fabsf(a.w - b.w));
  return fmaxf(m0, m1);
}

__device__ __forceinline__ float digammaf_dev(float x) {
  float r = 0.f;
  while (x < 6.f) { r -= 1.f / x; x += 1.f; }
  float inv = 1.f / x, inv2 = inv * inv;
  float s = logf(x) - 0.5f * inv
          - inv2 * (0.0833333333f - inv2 * (0.0083333333f - inv2 * 0.0039682540f));
  return r + s;
}

// top-4 insertion network: keeps m1>=m2>=m3>=m4 (4th largest ends in m4)
#define TOP4_PUSH(d)                                        \
  do {                                                      \
    float _t1 = fminf(m1, (d)); m1 = fmaxf(m1, (d));        \
    float _t2 = fminf(m2, _t1); m2 = fmaxf(m2, _t1);        \
    float _t3 = fminf(m3, _t2); m3 = fmaxf(m3, _t2);        \
    m4 = fmaxf(m4, _t3);                                    \
  } while (0)

// ---------- kernel 0: zero accumulators --------------------------------------
__global__ void k_init(double* acc) {
  if (threadIdx.x < 3) acc[threadIdx.x] = 0.0;
}

// ---------- kernel 1: 4th-largest joint Chebyshev distance per row -----------
__global__ __launch_bounds__(THREADS)
void k_knn(const float* __restrict__ X, const float* __restrict__ Y,
           float* __restrict__ knn, double* __restrict__ acc) {
  __shared__ __align__(16) float tX[WAVES][NBUF][TILE_J * DXD];   // 64KB
  __shared__ __align__(16) float tY[WAVES][NBUF][TILE_J * DYD];   // 64KB
  __shared__ float4 mrg[THREADS];
  __shared__ double red[ROWS];

  const int lane = threadIdx.x & 31;
  const int w    = threadIdx.x >> 5;
  const int rowbase = blockIdx.x * ROWS;
  const int row  = rowbase + lane;

  const float4* xr = (const float4*)(X + row * DXD);
  const float4* yr = (const float4*)(Y + row * DYD);
  float4 x0 = xr[0], x1 = xr[1], x2 = xr[2], x3 = xr[3];
  float4 y0 = yr[0], y1 = yr[1], y2 = yr[2], y3 = yr[3];

  float m1 = -INFINITY, m2 = -INFINITY, m3 = -INFINITY, m4 = -INFINITY;

  const int jbase0 = w * JCHUNK;
  // prime buffer 0
  wave_async_copy_4k(tX[w][0], X + (size_t)jbase0 * DXD, lane);
  wave_async_copy_4k(tY[w][0], Y + (size_t)jbase0 * DYD, lane);

  int buf = 0;
  for (int t = 0; t < TILES_PER_CHUNK; ++t) {
    if (t + 1 < TILES_PER_CHUNK) {
      const int jn = jbase0 + (t + 1) * TILE_J;
      wait_ds0();                                 // drain reads of buf^1 before overwrite
      wave_async_copy_4k(tX[w][buf ^ 1], X + (size_t)jn * DXD, lane);
      wave_async_copy_4k(tY[w][buf ^ 1], Y + (size_t)jn * DYD, lane);
      if (t + 2 < TILES_PER_CHUNK) {
        __builtin_prefetch(X + (size_t)(jn + TILE_J) * DXD, 0, 0);
        __builtin_prefetch(Y + (size_t)(jn + TILE_J) * DYD, 0, 0);
      }
      wait_async_prev();                          // tile t landed; t+1 still in flight
    } else {
      wait_async0();
    }

    const float4* px = (const float4*)tX[w][buf];
    const float4* py = (const float4*)tY[w][buf];
#pragma unroll 2
    for (int j = 0; j < TILE_J; ++j) {            // wave-uniform j -> LDS broadcast
      float d = linf4(x0, px[j * 4 + 0]);
      d = fmaxf(d, linf4(x1, px[j * 4 + 1]));
      d = fmaxf(d, linf4(x2, px[j * 4 + 2]));
      d = fmaxf(d, linf4(x3, px[j * 4 + 3]));
      d = fmaxf(d, linf4(y0, py[j * 4 + 0]));
      d = fmaxf(d, linf4(y1, py[j * 4 + 1]));
      d = fmaxf(d, linf4(y2, py[j * 4 + 2]));
      d = fmaxf(d, linf4(y3, py[j * 4 + 3]));
      TOP4_PUSH(d);
    }
    buf ^= 1;
  }

  mrg[threadIdx.x] = make_float4(m1, m2, m3, m4);
  __syncthreads();

  if (threadIdx.x < ROWS) {                       // merge the 8 per-wave top-4 sets
    float M1 = -INFINITY, M2 = -INFINITY, M3 = -INFINITY, M4 = -INFINITY;
#pragma unroll
    for (int ww = 0; ww < WAVES; ++ww) {
      float4 v = mrg[ww * 32 + threadIdx.x];
      float vv[4] = {v.x, v.y, v.z, v.w};
#pragma unroll
      for (int q = 0; q < 4; ++q) {
        float d = vv[q], m1 = M1, m2 = M2, m3 = M3, m4 = M4;
        TOP4_PUSH(d);
        M1 = m1; M2 = m2; M3 = m3; M4 = m4;
      }
    }
    knn[rowbase + threadIdx.x] = M4;
    red[threadIdx.x] = (double)logf(M4);
  }
  __syncthreads();
  if (threadIdx.x == 0) {
    double s = 0.0;
    for (int i = 0; i < ROWS; ++i) s += red[i];
    atomicAdd(&acc[0], s);
  }
}

// ---------- kernel 2: marginal ball counts + digamma sums --------------------
__global__ __launch_bounds__(THREADS)
void k_count(const float* __restrict__ X, const float* __restrict__ Y,
             const float* __restrict__ knn, double* __restrict__ acc) {
  __shared__ __align__(16) float tX[WAVES][NBUF][TILE_J * DXD];
  __shared__ __align__(16) float tY[WAVES][NBUF][TILE_J * DYD];
  __shared__ int2 cb[THREADS];
  __shared__ double rdx[ROWS], rdy[ROWS];

  const int lane = threadIdx.x & 31;
  const int w    = threadIdx.x >> 5;
  const int rowbase = blockIdx.x * ROWS;
  const int row  = rowbase + lane;

  const float4* xr = (const float4*)(X + row * DXD);
  const float4* yr = (const float4*)(Y + row * DYD);
  float4 x0 = xr[0], x1 = xr[1], x2 = xr[2], x3 = xr[3];
  float4 y0 = yr[0], y1 = yr[1], y2 = yr[2], y3 = yr[3];
  const float r = knn[row] - 1e-15f;              // radii = knn_dis - EPS

  int cx = 0, cy = 0;
  const int jbase0 = w * JCHUNK;
  wave_async_copy_4k(tX[w][0], X + (size_t)jbase0 * DXD, lane);
  wave_async_copy_4k(tY[w][0], Y + (size_t)jbase0 * DYD, lane);

  int buf = 0;
  for (int t = 0; t < TILES_PER_CHUNK; ++t) {
    if (t + 1 < TILES_PER_CHUNK) {
      const int jn = jbase0 + (t + 1) * TILE_J;
      wait_ds0();
      wave_async_copy_4k(tX[w][buf ^ 1], X + (size_t)jn * DXD, lane);
      wave_async_copy_4k(tY[w][buf ^ 1], Y + (size_t)jn * DYD, lane);
      if (t + 2 < TILES_PER_CHUNK) {
        __builtin_prefetch(X + (size_t)(jn + TILE_J) * DXD, 0, 0);
        __builtin_prefetch(Y + (size_t)(jn + TILE_J) * DYD, 0, 0);
      }
      wait_async_prev();
    } else {
      wait_async0();
    }

    const float4* px = (const float4*)tX[w][buf];
    const float4* py = (const float4*)tY[w][buf];
#pragma unroll 2
    for (int j = 0; j < TILE_J; ++j) {
      float dxm = linf4(x0, px[j * 4 + 0]);
      dxm = fmaxf(dxm, linf4(x1, px[j * 4 + 1]));
      dxm = fmaxf(dxm, linf4(x2, px[j * 4 + 2]));
      dxm = fmaxf(dxm, linf4(x3, px[j * 4 + 3]));
      float dym = linf4(y0, py[j * 4 + 0]);
      dym = fmaxf(dym, linf4(y1, py[j * 4 + 1]));
      dym = fmaxf(dym, linf4(y2, py[j * 4 + 2]));
      dym = fmaxf(dym, linf4(y3, py[j * 4 + 3]));
      cx += (dxm <= r) ? 1 : 0;
      cy += (dym <= r) ? 1 : 0;
    }
    buf ^= 1;
  }

  cb[threadIdx.x] = make_int2(cx, cy);
  __syncthreads();

  if (threadIdx.x < ROWS) {
    int nx = 0, ny = 0;
#pragma unroll
    for (int ww = 0; ww < WAVES; ++ww) {
      int2 v = cb[ww * 32 + threadIdx.x];
      nx += v.x; ny += v.y;
    }
    rdx[threadIdx.x] = (double)digammaf_dev((float)nx);
    rdy[threadIdx.x] = (double)digammaf_dev((float)ny);
  }
  __syncthreads();
  if (threadIdx.x == 0) {
    double sx = 0.0, sy = 0.0;
    for (int i = 0; i < ROWS; ++i) { sx += rdx[i]; sy += rdy[i]; }
    atomicAdd(&acc[1], sx);
    atomicAdd(&acc[2], sy);
  }
}

// ---------- kernel 3: combine ------------------------------------------------
__global__ void k_final(const double* acc, const int* kptr, float* out) {
  if (threadIdx.x == 0 && blockIdx.x == 0) {
    const float n    = (float)NPTS;
    const float mlr  = (float)(acc[0] / (double)NPTS);   // mean(log knn_dis)
    const float mdx  = (float)(acc[1] / (double)NPTS);   // mean(digamma(nx))
    const float mdy  = (float)(acc[2] / (double)NPTS);   // mean(digamma(ny))
    const float dign = digammaf_dev(n);
    const float digk = digammaf_dev((float)kptr[0]);
    const float l2   = logf(2.0f);
    const float dxy  = (float)(DXD + DYD);
    float ans_xy = -digk + dign + dxy * l2 + dxy * mlr;
    float ans_x  = dign + (float)DXD * l2 - mdx + (float)DXD * mlr;
    float ans_y  = dign + (float)DYD * l2 - mdy + (float)DYD * mlr;
    out[0] = ans_x + ans_y - ans_xy;
  }
}

// ---------- host entry -------------------------------------------------------
extern "C" void kernel_launch(void* const* d_in, const int* in_sizes, int n_in,
                              void* d_out, int out_size, void* d_ws, size_t ws_size,
                              hipStream_t stream) {
  const float* X  = (const float*)d_in[0];   // x_samples [8192,16] f32
  const float* Y  = (const float*)d_in[1];   // y_samples [8192,16] f32
  const int*   kp = (const int*)d_in[2];     // k (scalar)
  float* out = (float*)d_out;

  float*  knn = (float*)d_ws;                                  // N floats
  double* acc = (double*)((char*)d_ws + NPTS * sizeof(float)); // 3 doubles

  k_init <<<1, 32, 0, stream>>>(acc);
  k_knn  <<<NPTS / ROWS, THREADS, 0, stream>>>(X, Y, knn, acc);
  k_count<<<NPTS / ROWS, THREADS, 0, stream>>>(X, Y, knn, acc);
  k_final<<<1, 1, 0, stream>>>(acc, kp, out);
}